// CrossSpectralDensity_83820581748937
// MI455X (gfx1250) — compile-verified
//
#include <hip/hip_runtime.h>

#define B_DIM 32
#define T_DIM 4096
#define F_DIM 16
#define NFREQ 256
#define LAG_L 2047          // (T-1)/2
#define PAD   16            // leading zeros so s can start at -16
#define YPAD_LEN 6656       // >= PAD + T + lag_max(2063) + 240 + 1; multiple of 256
#define K_TOT 4128          // s = -16 .. 4111, multiple of 16 (4 waves x step 4)
#define NLAG_PAD 2304       // 9 tiles x 256 lags

typedef __attribute__((ext_vector_type(2))) float v2f;
typedef __attribute__((ext_vector_type(8))) float v8f;

// Kernel 1: ypad[b, PAD+t] = sum_c x[b,t,c]; zeros elsewhere (front pad + tail).
__global__ void csd_reduce_pad(const float* __restrict__ x,
                               float* __restrict__ ypad) {
    const int t = blockIdx.x * blockDim.x + threadIdx.x;   // 0..YPAD_LEN-1
    const int b = blockIdx.y;
    float v = 0.0f;
    if (t >= PAD && t < PAD + T_DIM) {
        const float4* xr = (const float4*)(x + ((size_t)b * T_DIM + (t - PAD)) * F_DIM);
        float4 p = xr[0], q = xr[1], r = xr[2], s = xr[3];
        v = (p.x + p.y + p.z + p.w) + (q.x + q.y + q.z + q.w) +
            (r.x + r.y + r.z + r.w) + (s.x + s.y + s.z + s.w);
    }
    ypad[(size_t)b * YPAD_LEN + t] = v;
}

// Kernel 2: autocorrelation via WMMA.
// D[mi,ni] = sum_kk ys[kk+mi] * ys[kk+lam+16*ni]  ->  a[lag = lam + 16*ni - mi].
// One block per (batch, lag-tile); 9 tiles cover lags 0..2303. Padding makes the
// linear autocorrelation exact. 4 waves split K=4128; LDS reduction at the end.
__global__ void csd_autocorr_wmma(const float* __restrict__ ypad,
                                  float* __restrict__ a) {
    __shared__ float ys[YPAD_LEN];
    __shared__ float red[4][32][8];
    const int b   = blockIdx.y;
    const int lam = 15 + 256 * blockIdx.x;                 // tile lag base + 15
    const int tid = threadIdx.x;

    const float* yb = ypad + (size_t)b * YPAD_LEN;
    for (int i = tid; i < YPAD_LEN; i += 128) ys[i] = yb[i];
    __syncthreads();

    const int lane = tid & 31;
    const int wave = __builtin_amdgcn_readfirstlane(tid >> 5);   // scalar -> uniform loop
    const int l15  = lane & 15;
    const int koff = (lane >> 4) * 2;                      // K = {0,1} | {2,3}
    const int aoff = l15;                                  // A: mi = lane&15
    const int boff = lam + 16 * l15;                       // B: lam + 16*ni

    v8f acc = {};
    const int kstart = wave * (K_TOT / 4);                 // 1032 K per wave
    #pragma unroll 2
    for (int k = kstart; k < kstart + K_TOT / 4; k += 4) {
        const int kk = k + koff;
        v2f av; av[0] = ys[kk + aoff]; av[1] = ys[kk + 1 + aoff];
        v2f bv; bv[0] = ys[kk + boff]; bv[1] = ys[kk + 1 + boff];
        acc = __builtin_amdgcn_wmma_f32_16x16x4_f32(
            false, av, false, bv, (short)0, acc, false, false);
    }

    #pragma unroll
    for (int j = 0; j < 8; ++j) red[wave][lane][j] = acc[j];
    __syncthreads();

    if (tid < 32) {
        const int M0 = (tid >> 4) * 8;                     // D: lanes>=16 hold M=8..15
        const int N  = tid & 15;
        #pragma unroll
        for (int j = 0; j < 8; ++j) {
            float s = red[0][tid][j] + red[1][tid][j] +
                      red[2][tid][j] + red[3][tid][j];
            int lag = lam + 16 * N - (M0 + j);             // unique in [256*tile, 256*tile+255]
            a[(size_t)b * NLAG_PAD + lag] = s;
        }
    }
}

// Kernel 3: c[b,t] = a[b, |t - 2047|]  (lag window -2047..2048; autocorr symmetric)
__global__ void csd_expand(const float* __restrict__ a,
                           float* __restrict__ c) {
    const int t = blockIdx.x * blockDim.x + threadIdx.x;
    const int b = blockIdx.y;
    int m = t - LAG_L; if (m < 0) m = -m;
    c[(size_t)b * T_DIM + t] = a[(size_t)b * NLAG_PAD + m];
}

// Kernel 4: out[b,f] = (1/16384) * sum_t c[b,t] * e^{-2*pi*i*f*t/4096}, f=0..255
// GEMM pair via V_WMMA_F32_16X16X4_F32; twiddles from exact integer phase.
__global__ void csd_dft_wmma(const float* __restrict__ c,
                             float* __restrict__ out) {
    __shared__ float red_re[4][32][8];
    __shared__ float red_im[4][32][8];

    const int lane = threadIdx.x & 31;
    const int wave = __builtin_amdgcn_readfirstlane((int)(threadIdx.x >> 5));
    const int m0 = blockIdx.y * 16;                        // batch tile
    const int n0 = blockIdx.x * 16;                        // frequency tile

    const int l15  = lane & 15;
    const int koff = (lane >> 4) * 2;
    const int row  = m0 + l15;
    const int f    = n0 + l15;

    const float2* crow = (const float2*)(c + (size_t)row * T_DIM);
    const float w = -6.28318530717958647692f / (float)T_DIM;

    v8f acc_re = {};
    v8f acc_im = {};

    const int kstart = wave * (T_DIM / 4);                 // 1024 K per wave
    #pragma unroll 4
    for (int k = kstart; k < kstart + T_DIM / 4; k += 4) {
        const int kk = k + koff;
        float2 av = crow[kk >> 1];
        v2f av2; av2[0] = av.x; av2[1] = av.y;

        int p0 = (f * kk)       & (T_DIM - 1);             // exact phase mod 4096
        int p1 = (f * (kk + 1)) & (T_DIM - 1);
        float s0, c0, s1, c1;
        __sincosf(w * (float)p0, &s0, &c0);
        __sincosf(w * (float)p1, &s1, &c1);
        v2f bre; bre[0] = c0; bre[1] = c1;
        v2f bim; bim[0] = s0; bim[1] = s1;

        acc_re = __builtin_amdgcn_wmma_f32_16x16x4_f32(
            false, av2, false, bre, (short)0, acc_re, false, false);
        acc_im = __builtin_amdgcn_wmma_f32_16x16x4_f32(
            false, av2, false, bim, (short)0, acc_im, false, false);
    }

    #pragma unroll
    for (int j = 0; j < 8; ++j) {
        red_re[wave][lane][j] = acc_re[j];
        red_im[wave][lane][j] = acc_im[j];
    }
    __syncthreads();

    if (threadIdx.x < 32) {
        const float scale = 1.0f / 16384.0f;               // 1/(sqrt(4096)*256)
        const int ln = threadIdx.x;
        const int Mbase = (ln >> 4) * 8;
        const int N = ln & 15;
        #pragma unroll
        for (int j = 0; j < 8; ++j) {
            float sr = red_re[0][ln][j] + red_re[1][ln][j] +
                       red_re[2][ln][j] + red_re[3][ln][j];
            float si = red_im[0][ln][j] + red_im[1][ln][j] +
                       red_im[2][ln][j] + red_im[3][ln][j];
            int bb = m0 + Mbase + j;
            int ff = n0 + N;
            out[((size_t)bb * NFREQ + ff) * 2 + 0] = sr * scale;
            out[((size_t)bb * NFREQ + ff) * 2 + 1] = si * scale;
        }
    }
}

extern "C" void kernel_launch(void* const* d_in, const int* in_sizes, int n_in,
                              void* d_out, int out_size, void* d_ws, size_t ws_size,
                              hipStream_t stream) {
    const float* x = (const float*)d_in[0];
    float* out = (float*)d_out;

    float* ypad = (float*)d_ws;                       // 32*6656 floats (852KB)
    float* a    = ypad + (size_t)B_DIM * YPAD_LEN;    // 32*2304 floats (288KB)
    float* c    = a    + (size_t)B_DIM * NLAG_PAD;    // 32*4096 floats (512KB)

    csd_reduce_pad   <<<dim3(YPAD_LEN / 256, B_DIM), 256, 0, stream>>>(x, ypad);
    csd_autocorr_wmma<<<dim3(NLAG_PAD / 256, B_DIM), 128, 0, stream>>>(ypad, a);
    csd_expand       <<<dim3(T_DIM / 256, B_DIM),    256, 0, stream>>>(a, c);
    csd_dft_wmma     <<<dim3(NFREQ / 16, B_DIM / 16), 128, 0, stream>>>(c, out);
}